// FlameKNN_11295763988791
// MI455X (gfx1250) — compile-verified
//
#include <hip/hip_runtime.h>
#include <hip/hip_bf16.h>

typedef __attribute__((ext_vector_type(2))) float v2f;
typedef __attribute__((ext_vector_type(4))) float v4f;
typedef __attribute__((ext_vector_type(8))) float v8f;

#define N_G      100000      // means (queries)
#define N_V      5023        // vertices (database)
#define KNN      8
#define TOPK     9           // reference keeps K+1
#define TILE_V   16
#define N_TILES  314         // ceil(5023/16)
#define NV_PAD   (N_TILES*TILE_V)   // 5024, padded with far-away sentinels
#define WAVES    8
#define BLK_THREADS (WAVES*32)
#define MEANS_PER_WAVE 16
#define MEANS_PER_BLOCK (WAVES*MEANS_PER_WAVE)  // 128
#define SCR_FLOATS 576       // per-wave scratch (tile transpose / list merge)
#define COLSTRIDE  20        // 80B column stride: 16B aligned, conflict-free selection loads

#define LDS_FLOATS (NV_PAD*4 + WAVES*SCR_FLOATS)

// Insert (dd, vi) into the private sorted top-9 (bd ascending, bd[8] = worst).
#define INSERT9(dd, vi)                                                   \
    if ((dd) < bd[TOPK-1]) {                                              \
        bd[TOPK-1] = (dd); bi[TOPK-1] = (vi);                             \
        _Pragma("unroll")                                                 \
        for (int _i = TOPK-1; _i > 0; --_i) {                             \
            if (bd[_i] < bd[_i-1]) {                                      \
                float _td = bd[_i]; bd[_i] = bd[_i-1]; bd[_i-1] = _td;    \
                int   _ti = bi[_i]; bi[_i] = bi[_i-1]; bi[_i-1] = _ti;    \
            }                                                             \
        }                                                                 \
    }

__global__ __launch_bounds__(BLK_THREADS)
void flame_knn_wmma(const float* __restrict__ means,
                    const float* __restrict__ verts,
                    float* __restrict__ out)
{
    extern __shared__ float smem[];
    float* s_vert = smem;              // [NV_PAD*4] : x, y, z, |v|^2
    float* s_scr  = smem + NV_PAD*4;   // [WAVES*SCR_FLOATS]

    const int tid = threadIdx.x;

    // Stage vertices (+ squared norms) into LDS; pad with huge-score sentinels.
    for (int j = tid; j < NV_PAD; j += BLK_THREADS) {
        float x = 0.f, y = 0.f, z = 0.f, sq = 3.0e37f;
        if (j < N_V) {
            x = verts[3*j+0]; y = verts[3*j+1]; z = verts[3*j+2];
            sq = x*x + y*y + z*z;
        }
        s_vert[4*j+0] = x; s_vert[4*j+1] = y;
        s_vert[4*j+2] = z; s_vert[4*j+3] = sq;
    }
    __syncthreads();

    const int wave = tid >> 5;       // wave32
    const int lane = tid & 31;
    const int half = lane >> 4;      // 0: lanes 0-15, 1: lanes 16-31
    const int l16  = lane & 15;

    float* wscr = s_scr + wave*SCR_FLOATS;

    // Each wave owns 16 means; lane pair (l16, l16+16) shares mean row l16.
    const int  mrow   = blockIdx.x*MEANS_PER_BLOCK + wave*MEANS_PER_WAVE + l16;
    const bool mvalid = (mrow < N_G);
    float mx = 0.f, my = 0.f, mz = 0.f;
    if (mvalid) { mx = means[3*mrow+0]; my = means[3*mrow+1]; mz = means[3*mrow+2]; }

    // Ranking score: s = |v|^2 - 2 m.v   (|m|^2 dropped: constant per row, cancels in d9-d8)
    // A (16x4): rows (-2mx, -2my, -2mz, 1); B (4x16): cols (vx, vy, vz, |v|^2); C = 0 (inline).
    v2f afrag;
    afrag.x = half ? (-2.f*mz) : (-2.f*mx);
    afrag.y = half ? 1.0f      : (-2.f*my);

    // Private sorted top-9 (ascending).
    float bd[TOPK];
    int   bi[TOPK];
#pragma unroll
    for (int i = 0; i < TOPK; ++i) { bd[i] = 3.0e38f; bi[i] = 0; }

    for (int t = 0; t < N_TILES; ++t) {
        const int vbase = t*TILE_V;
        const float* vp = &s_vert[4*(vbase + l16)];

        // One uniform 8-byte LDS load per lane: half 0 -> (vx,vy), half 1 -> (vz,|v|^2).
        const v2f bfrag = *(const v2f*)(vp + 2*half);

        v8f cf = {};   // lowers to inline-0 SRC2
        v8f d = __builtin_amdgcn_wmma_f32_16x16x4_f32(
            false, afrag, false, bfrag, (short)0, cf, false, false);

        // Scatter D tile to LDS: element (m = v + 8*half, n = l16) -> wscr[n*COLSTRIDE + m]
        {
            float* tp = wscr + l16*COLSTRIDE + half*8;
            v4f lo, hi;
#pragma unroll
            for (int v = 0; v < 4; ++v) { lo[v] = d[v]; hi[v] = d[v+4]; }
            *(v4f*)(tp)     = lo;
            *(v4f*)(tp + 4) = hi;
        }
        __builtin_amdgcn_wave_barrier();

        // Transpose-read: lane handles mean m = l16, vertex cols n = 8*half + k.
        // Batch all 8 loads (single dscnt wait), then group-reject via a min tree.
        float dd[8];
#pragma unroll
        for (int k = 0; k < 8; ++k)
            dd[k] = wscr[(8*half + k)*COLSTRIDE + l16];

        const float mn = fminf(fminf(fminf(dd[0], dd[1]), fminf(dd[2], dd[3])),
                               fminf(fminf(dd[4], dd[5]), fminf(dd[6], dd[7])));
        if (mn < bd[TOPK-1]) {
#pragma unroll
            for (int k = 0; k < 8; ++k) {
                const int vi = vbase + 8*half + k;
                INSERT9(dd[k], vi)
            }
        }
        __builtin_amdgcn_wave_barrier();   // keep next tile's stores after this tile's reads
    }

    // Merge the two half-row lists: lanes 16-31 publish, lanes 0-15 absorb.
    int* wscri = (int*)wscr;
    if (half) {
#pragma unroll
        for (int k = 0; k < TOPK; ++k) {
            wscr [l16*TOPK + k]           = bd[k];
            wscri[16*TOPK + l16*TOPK + k] = bi[k];
        }
    }
    __builtin_amdgcn_wave_barrier();
    asm volatile("" ::: "memory");
    if (!half) {
#pragma unroll
        for (int k = 0; k < TOPK; ++k) {
            const float dv = wscr [l16*TOPK + k];
            const int   vi = wscri[16*TOPK + l16*TOPK + k];
            INSERT9(dv, vi)
        }
        if (mvalid) {
            float* out_idx = out;                       // [N_G*KNN] indices (as float)
            float* out_rec = out + (size_t)N_G*KNN;     // [N_G] d9 - d8
            v4f i0, i1;
#pragma unroll
            for (int k = 0; k < 4; ++k) { i0[k] = (float)bi[k]; i1[k] = (float)bi[k+4]; }
            *(v4f*)&out_idx[(size_t)mrow*KNN]     = i0;
            *(v4f*)&out_idx[(size_t)mrow*KNN + 4] = i1;
            out_rec[mrow] = bd[TOPK-1] - bd[TOPK-2];    // |m|^2 cancels exactly in the reference too
        }
    }
    if (blockIdx.x == 0 && tid == 0)
        out[(size_t)N_G*KNN + N_G] = 0.0f;              // hit_rate
}

extern "C" void kernel_launch(void* const* d_in, const int* in_sizes, int n_in,
                              void* d_out, int out_size, void* d_ws, size_t ws_size,
                              hipStream_t stream) {
    (void)in_sizes; (void)n_in; (void)out_size; (void)d_ws; (void)ws_size;
    const float* means = (const float*)d_in[0];
    const float* verts = (const float*)d_in[1];
    float* out = (float*)d_out;

    const int blocks = (N_G + MEANS_PER_BLOCK - 1) / MEANS_PER_BLOCK;  // 782
    const size_t shmem = (size_t)LDS_FLOATS * sizeof(float);           // ~98.8 KB
    flame_knn_wmma<<<blocks, BLK_THREADS, shmem, stream>>>(means, verts, out);
}